// SimpleCA_86096914415705
// MI455X (gfx1250) — compile-verified
//
#include <hip/hip_runtime.h>

// NCA step for MI455X (gfx1250, wave32).
// Memory-bound: ~640 MiB traffic (1.25x input halo) -> ~27us floor @ 23.3 TB/s.
// 4->6 pointwise conv via V_WMMA_F32_16X16X4_F32 (K=4 exact fit).
// Tile fill via GLOBAL_LOAD_ASYNC_TO_LDS_B128 (ASYNCcnt) when available.

typedef __attribute__((ext_vector_type(2))) float v2f;
typedef __attribute__((ext_vector_type(4))) float v4f;
typedef __attribute__((ext_vector_type(8))) float v8f;
typedef __attribute__((ext_vector_type(4))) int   v4i;

#define BATCH 64
#define H 512
#define W 512
#define TILE_H 8
#define NTHREADS 256
#define NWAVES 8
#define PIX_PER_WAVE (TILE_H * W / NWAVES)   // 512
#define NCHUNK (PIX_PER_WAVE / 32)           // 16

#ifndef __has_builtin
#define __has_builtin(x) 0
#endif

#if __has_builtin(__builtin_amdgcn_global_load_async_to_lds_b128) && \
    __has_builtin(__builtin_amdgcn_s_wait_asynccnt)
#define USE_ASYNC_LDS 1
#else
#define USE_ASYNC_LDS 0
#endif

typedef __attribute__((address_space(1))) v4i g_v4i;   // global int4
typedef __attribute__((address_space(3))) v4i l_v4i;   // LDS int4

__global__ __launch_bounds__(NTHREADS)
void nca_step_kernel(const float* __restrict__ in,
                     const float* __restrict__ rnd,
                     const float* __restrict__ w1k,   // (4,6) row-major
                     const float* __restrict__ w1b,   // (6)
                     const float* __restrict__ w2k,   // (6,4) row-major
                     float* __restrict__ out)
{
    // Input tile: (TILE_H+2) rows x W pixels x float4 = 10*512*16 = 80 KB
    __shared__ float tile[(TILE_H + 2) * W * 4];
    // Per-wave h scratch: 32 pixels x 8 floats (6 used, padded) = 8 KB
    __shared__ float hbuf[NWAVES * 32 * 8];

    const int tid  = threadIdx.x;
    const int lane = tid & 31;
    const int wave = tid >> 5;

    const int blocks_per_img = H / TILE_H;           // 64
    const int b    = blockIdx.x / blocks_per_img;
    const int ty   = blockIdx.x % blocks_per_img;
    const int row0 = ty * TILE_H;

    const v4f* in4  = (const v4f*)in;
    v4f*       out4 = (v4f*)out;
    v4f*       t4   = (v4f*)tile;

    // ---- cooperative tile fill, toroidal in H ----
    for (int i = tid; i < (TILE_H + 2) * W; i += NTHREADS) {
        const int r  = i >> 9;                 // i / W
        const int x  = i & (W - 1);
        const int gy = (row0 - 1 + r + H) & (H - 1);
        const v4f* src = &in4[(size_t)b * H * W + (size_t)gy * W + x];
#if USE_ASYNC_LDS
        // one-instruction memory->LDS transfer, tracked by ASYNCcnt
        __builtin_amdgcn_global_load_async_to_lds_b128(
            (g_v4i*)src, (l_v4i*)&t4[i], /*offset=*/0, /*cpol=*/0);
#else
        t4[i] = *src;
#endif
    }
#if USE_ASYNC_LDS
    __builtin_amdgcn_s_wait_asynccnt(0);
#endif
    __syncthreads();

    // ---- WMMA B-matrix (w1, zero-padded 4x16) and bias-as-C, per ISA layout:
    // B v0: lanes 0-15 -> (K=0, N=lane), lanes 16-31 -> (K=2, N=lane-16); v1: K=1/K=3.
    // C/D: VGPR j, lanes 0-15 -> (M=j, N=lane); lanes 16-31 -> (M=j+8, N=lane-16).
    const int  n  = lane & 15;
    const bool lo = lane < 16;
    float wb0 = 0.f, wb1 = 0.f, cb = 0.f;
    if (n < 6) {
        const int k0 = lo ? 0 : 2;
        wb0 = w1k[k0 * 6 + n];
        wb1 = w1k[(k0 + 1) * 6 + n];
        cb  = w1b[n];
    }
    v2f Bm; Bm[0] = wb0; Bm[1] = wb1;

    // w2 is uniform -> scalar loads
    float w2s[24];
    #pragma unroll
    for (int i = 0; i < 24; ++i) w2s[i] = w2k[i];

    // ---- each wave: one tile row (512 px) as 16 chunks of 32 consecutive px ----
    for (int chunk = 0; chunk < NCHUNK; ++chunk) {
        const int p  = wave * PIX_PER_WAVE + chunk * 32 + lane;  // 0..4095
        const int ly = p >> 9;                                   // row in tile
        const int x  = p & (W - 1);
        const int tr = ly + 1;                                   // +1 halo
        const int xm = (x == 0) ? (W - 1) : x - 1;
        const int xp = (x == W - 1) ? 0 : x + 1;

        const v4f NW = t4[(tr - 1) * W + xm];
        const v4f Nc = t4[(tr - 1) * W + x ];
        const v4f NE = t4[(tr - 1) * W + xp];
        const v4f Wc = t4[ tr      * W + xm];
        const v4f Cc = t4[ tr      * W + x ];
        const v4f Ec = t4[ tr      * W + xp];
        const v4f SW = t4[(tr + 1) * W + xm];
        const v4f Sc = t4[(tr + 1) * W + x ];
        const v4f SE = t4[(tr + 1) * W + xp];

        // Perception (depthwise, channel i uses filter i)
        const float p0 = Cc.x;                                                // identity
        const float p1 = (NE.y - NW.y) + 2.f * (Ec.y - Wc.y) + (SE.y - SW.y); // sobel_x
        const float p2 = (SW.z - NW.z) + 2.f * (Sc.z - Nc.z) + (SE.z - NE.z); // sobel_y
        const float p3 = (NW.w + NE.w + SW.w + SE.w)
                       + 2.f * (Nc.w + Wc.w + Ec.w + Sc.w) - 12.f * Cc.w;     // laplacian

        float hv[6];
#if __has_builtin(__builtin_amdgcn_wmma_f32_16x16x4_f32)
        // h = relu(percep(16x4) @ w1(4x16) + bias) via two WMMA tiles.
        // A: v0 = lanes0-15:(M=lane,K=0) / lanes16-31:(M=lane-16,K=2); v1 = K=1/K=3.
        #pragma unroll
        for (int t = 0; t < 2; ++t) {
            const int src = t * 16 + n;
            const float s0 = __shfl(p0, src, 32);
            const float s1 = __shfl(p1, src, 32);
            const float s2 = __shfl(p2, src, 32);
            const float s3 = __shfl(p3, src, 32);
            v2f Am; Am[0] = lo ? s0 : s2; Am[1] = lo ? s1 : s3;
            v8f Cm;
            #pragma unroll
            for (int j = 0; j < 8; ++j) Cm[j] = cb;
            v8f D = __builtin_amdgcn_wmma_f32_16x16x4_f32(
                false, Am, false, Bm, (short)0, Cm, false, false);
            // relu + scatter h into wave-private LDS (pixel-major, stride 8)
            if (n < 6) {
                const int mbase = lo ? 0 : 8;
                #pragma unroll
                for (int j = 0; j < 8; ++j) {
                    hbuf[(wave * 32 + t * 16 + mbase + j) * 8 + n] = fmaxf(D[j], 0.0f);
                }
            }
        }
        // Gather this lane's pixel h[0..5]; same-wave DS ordering, no barrier.
        {
            const float* hp = &hbuf[(wave * 32 + lane) * 8];
            #pragma unroll
            for (int d = 0; d < 6; ++d) hv[d] = hp[d];
        }
#else
        // Scalar fallback (keeps compile green if builtin is absent)
        #pragma unroll
        for (int d = 0; d < 6; ++d) {
            float a = p0 * w1k[d] + p1 * w1k[6 + d] + p2 * w1k[12 + d]
                    + p3 * w1k[18 + d] + w1b[d];
            hv[d] = fmaxf(a, 0.0f);
        }
#endif
        // dx = h @ w2 (6->4), per-lane FMAs with uniform weights
        float dx0 = 0.f, dx1 = 0.f, dx2 = 0.f, dx3 = 0.f;
        #pragma unroll
        for (int d = 0; d < 6; ++d) {
            dx0 += hv[d] * w2s[d * 4 + 0];
            dx1 += hv[d] * w2s[d * 4 + 1];
            dx2 += hv[d] * w2s[d * 4 + 2];
            dx3 += hv[d] * w2s[d * 4 + 3];
        }

        const size_t gidx = (size_t)b * H * W + (size_t)(row0 + ly) * W + x;
        // rand: read-once -> non-temporal; out: write-once -> non-temporal
        const float rv = __builtin_nontemporal_load(&rnd[gidx]);
        const float m  = floorf(rv + 0.5f);          // stochastic update mask
        v4f o;
        o.x = Cc.x + dx0 * m;
        o.y = Cc.y + dx1 * m;
        o.z = Cc.z + dx2 * m;
        o.w = Cc.w + dx3 * m;
        __builtin_nontemporal_store(o, &out4[gidx]); // B128 NT store
    }
}

extern "C" void kernel_launch(void* const* d_in, const int* in_sizes, int n_in,
                              void* d_out, int out_size, void* d_ws, size_t ws_size,
                              hipStream_t stream) {
    (void)in_sizes; (void)n_in; (void)out_size; (void)d_ws; (void)ws_size;
    const float* in  = (const float*)d_in[0];
    const float* rnd = (const float*)d_in[1];
    const float* w1k = (const float*)d_in[2];
    const float* w1b = (const float*)d_in[3];
    const float* w2k = (const float*)d_in[4];
    float* out = (float*)d_out;

    dim3 grid(BATCH * (H / TILE_H));   // 64 * 64 = 4096 blocks
    dim3 block(NTHREADS);
    nca_step_kernel<<<grid, block, 0, stream>>>(in, rnd, w1k, w1b, w2k, out);
}